// Memory_42056319762659
// MI455X (gfx1250) — compile-verified
//
#include <hip/hip_runtime.h>
#include <math.h>

// Problem constants (from the reference)
#define B_   2
#define H_   64
#define W_   64
#define CK_  256
#define CV_  3
#define M_   256
#define KER_ 7
#define P_   49   // KER*KER
#define HW_  4096 // H*W

#define ZPAD_FLOATS 512  // zeroed scratch page for padded patch rows

typedef __attribute__((ext_vector_type(2))) float v2f;
typedef __attribute__((ext_vector_type(4))) float v4f;
typedef __attribute__((ext_vector_type(8))) float v8f;

// -----------------------------------------------------------------------------
// Kernel 0: zero the pad page (d_ws is poisoned; zero-pad loads read from here).
// -----------------------------------------------------------------------------
__global__ void __launch_bounds__(ZPAD_FLOATS)
zero_pad_kernel(float* __restrict__ zbuf) {
  zbuf[threadIdx.x] = 0.0f;
}

// -----------------------------------------------------------------------------
// Kernel 1: fused implicit-im2col fp32 WMMA GEMM + online softmax row stats.
//   logits l[n,m] = sum_{p,c} k_patch[n, p*CK+c] * m_k[m, p, c]
//   shw[n] = max_m softmax(l[n,:]) = 1 / sum_m exp(l[n,m] - max_m l[n,m])
// Block = 64 threads (2 waves). Each block owns one tile of 16 consecutive n
// (same image row h, w0..w0+15). Wave 0 covers m-tiles 0..7 (m=0..127),
// wave 1 covers m-tiles 8..15 (m=128..255); merged via LDS at the end.
//
// K mapping trick: per 8-channel chunk, one b128 load/lane feeds two WMMAs.
//   lo-WMMA: K{0,1}=c0+{0,1} (lanes 0-15), K{2,3}=c0+{4,5} (lanes 16-31)
//   hi-WMMA: K{0,1}=c0+{2,3},              K{2,3}=c0+{6,7}
// A and B use the same permutation, so the K-sum is unchanged.
// Zero padding: invalid lanes read from the zeroed pad page (no exec dance).
// -----------------------------------------------------------------------------
__global__ void __launch_bounds__(64)
gemm_rowstats_kernel(const float* __restrict__ k,
                     const float* __restrict__ mk,
                     const float* __restrict__ zbuf,
                     float* __restrict__ shw) {
  const int wave  = threadIdx.x >> 5;   // 0..1  -> which half of m
  const int lane  = threadIdx.x & 31;
  const int mrow  = lane & 15;          // A: n-row in tile / B: m-col in tile
  const int khalf = lane >> 4;          // lanes 0-15 vs 16-31 K split

  const int nt = blockIdx.x & (HW_ / 16 - 1);  // n-tile 0..255
  const int b  = blockIdx.x / (HW_ / 16);
  const int h  = nt >> 2;               // 4 tiles per image row (W=64)
  const int w0 = (nt & 3) * 16;

  const float* kb  = k  + (size_t)b * HW_ * CK_;
  const float* mkb = mk + (size_t)b * M_ * P_ * CK_
                        + (size_t)(wave * 8 * 16) * P_ * CK_   // this wave's m base
                        + (size_t)mrow * P_ * CK_ + 4 * khalf;
  const float* zrow = zbuf + 4 * khalf;

  v8f acc[8];
  const v8f vzero = {0.f, 0.f, 0.f, 0.f, 0.f, 0.f, 0.f, 0.f};
#pragma unroll
  for (int t = 0; t < 8; ++t) acc[t] = vzero;

  for (int p = 0; p < P_; ++p) {
    const int kh = p / KER_;
    const int kw = p - kh * KER_;
    const int hh = h + kh - (KER_ / 2);
    const int ww = w0 + mrow + kw - (KER_ / 2);
    const bool valid = (hh >= 0) & (hh < H_) & (ww >= 0) & (ww < W_);
    // Unconditional loads: invalid (padded) lanes read zeros from the pad page.
    const float* arow = valid ? (kb + (size_t)(hh * W_ + ww) * CK_ + 4 * khalf)
                              : zrow;
    const float* brow = mkb + (size_t)p * CK_;

    // Speculative prefetch of next p's A row (silently dropped if bogus).
    {
      const int kh2 = (p + 1) / KER_;
      const int kw2 = (p + 1) - kh2 * KER_;
      const int hh2 = h + kh2 - (KER_ / 2);
      const int ww2 = w0 + mrow + kw2 - (KER_ / 2);
      __builtin_prefetch(kb + (size_t)(hh2 * W_ + ww2) * CK_, 0, 1);
    }

    for (int c0 = 0; c0 < CK_; c0 += 8) {
      const v4f a = *(const v4f*)(arow + c0);
      v4f bf[8];
#pragma unroll
      for (int t = 0; t < 8; ++t)
        bf[t] = *(const v4f*)(brow + (size_t)t * 16 * P_ * CK_ + c0);

      const v2f alo = {a.x, a.y};
      const v2f ahi = {a.z, a.w};
#pragma unroll
      for (int t = 0; t < 8; ++t) {
        const v2f blo = {bf[t].x, bf[t].y};
        acc[t] = __builtin_amdgcn_wmma_f32_16x16x4_f32(
            false, alo, false, blo, (short)0, acc[t], false, false);
      }
#pragma unroll
      for (int t = 0; t < 8; ++t) {
        const v2f bhi = {bf[t].z, bf[t].w};
        acc[t] = __builtin_amdgcn_wmma_f32_16x16x4_f32(
            false, ahi, false, bhi, (short)0, acc[t], false, false);
      }
    }
  }

  // C layout: acc[t][i] at lane L = l[n = n0 + (L>>4)*8 + i, m = wave*128 + t*16 + (L&15)]
  // Per-row max over this wave's 128 m's: reduce over t in-register, then over
  // the 16 lanes of the half (shfl_xor 1,2,4,8 stays inside each 16-lane group).
  float rmax[8], rsum[8];
#pragma unroll
  for (int i = 0; i < 8; ++i) {
    float mx = acc[0][i];
#pragma unroll
    for (int t = 1; t < 8; ++t) mx = fmaxf(mx, acc[t][i]);
#pragma unroll
    for (int d = 1; d < 16; d <<= 1) mx = fmaxf(mx, __shfl_xor(mx, d, 32));
    rmax[i] = mx;
  }
#pragma unroll
  for (int i = 0; i < 8; ++i) {
    float s = 0.f;
#pragma unroll
    for (int t = 0; t < 8; ++t) s += __expf(acc[t][i] - rmax[i]);
#pragma unroll
    for (int d = 1; d < 16; d <<= 1) s += __shfl_xor(s, d, 32);
    rsum[i] = s;
  }

  // Merge the two m-halves with the standard (max, Z) softmax merge via LDS.
  __shared__ float lm[2][16];
  __shared__ float lz[2][16];
  if (mrow == 0) {
#pragma unroll
    for (int i = 0; i < 8; ++i) {
      const int r = khalf * 8 + i;
      lm[wave][r] = rmax[i];
      lz[wave][r] = rsum[i];
    }
  }
  __syncthreads();
  if (threadIdx.x < 16) {
    const int r = threadIdx.x;
    const float m0 = lm[0][r], m1 = lm[1][r];
    const float mm = fmaxf(m0, m1);
    const float Z  = lz[0][r] * __expf(m0 - mm) + lz[1][r] * __expf(m1 - mm);
    shw[b * HW_ + nt * 16 + r] = 1.0f / Z;   // == max_m softmax(l[n,:])
  }
}

// -----------------------------------------------------------------------------
// Kernel 2: stable rank selection. rank(n) = #{j: s[j]<s[n] or (s[j]==s[n] && j<n)}
// == position of n in jnp.argsort(s) (stable, ascending). Ranks < M scatter.
// Data is 16KB/batch -> lives in L2; 4096 threads x 4096 scans is trivial.
// -----------------------------------------------------------------------------
__global__ void __launch_bounds__(256)
rank_select_kernel(const float* __restrict__ shw, int* __restrict__ sel) {
  const int b = blockIdx.y;
  const int n = blockIdx.x * 256 + threadIdx.x;
  const float* s = shw + b * HW_;
  const float sn = s[n];
  int rank = 0;
  for (int j = 0; j < HW_; ++j) {
    const float sj = s[j];
    rank += (sj < sn) || ((sj == sn) && (j < n));
  }
  if (rank < M_) sel[b * M_ + rank] = n;
}

// -----------------------------------------------------------------------------
// Kernel 3: gather selected patch rows directly from k/v (zero-padded im2col
// on the fly). m_k_new = k_patch[sel] as [B,M,P,CK]; m_v_new likewise [B,M,P,CV].
// -----------------------------------------------------------------------------
__global__ void __launch_bounds__(256)
gather_kernel(const float* __restrict__ k, const float* __restrict__ v,
              const int* __restrict__ sel,
              float* __restrict__ mk_out, float* __restrict__ mv_out) {
  const int p = blockIdx.x;
  const int r = blockIdx.y;
  const int b = blockIdx.z;
  const int c = threadIdx.x;

  const int n  = sel[b * M_ + r];
  const int h  = n >> 6;
  const int w  = n & 63;
  const int kh = p / KER_;
  const int kw = p - kh * KER_;
  const int hh = h + kh - (KER_ / 2);
  const int ww = w + kw - (KER_ / 2);
  const bool ok = (hh >= 0) & (hh < H_) & (ww >= 0) & (ww < W_);
  const int src = (hh << 6) + ww;

  const float kv = ok ? k[((size_t)(b * HW_ + src) << 8) + c] : 0.0f;
  mk_out[(((size_t)(b * M_ + r) * P_ + p) << 8) + c] = kv;

  if (c < CV_) {
    const float vv = ok ? v[(size_t)(b * HW_ + src) * CV_ + c] : 0.0f;
    mv_out[((size_t)(b * M_ + r) * P_ + p) * CV_ + c] = vv;
  }
}

// -----------------------------------------------------------------------------
// Launch. Inputs (setup_inputs order): 0:k 1:v 2:rkn_score 3:m_k 4:m_v 5:m_u.
// rkn_score / m_v / m_u are provably dead (wv_bool is always true: THRESH=49
// while max of softmax <= 1, so m_k_new = write_k, m_v_new = write_v).
// d_out = m_k_new flat (B*M*P*CK) followed by m_v_new flat (B*M*P*CV).
// -----------------------------------------------------------------------------
extern "C" void kernel_launch(void* const* d_in, const int* in_sizes, int n_in,
                              void* d_out, int out_size, void* d_ws, size_t ws_size,
                              hipStream_t stream) {
  const float* k  = (const float*)d_in[0];
  const float* v  = (const float*)d_in[1];
  const float* mk = (const float*)d_in[3];

  float* out    = (float*)d_out;
  float* mk_out = out;
  float* mv_out = out + (size_t)B_ * M_ * P_ * CK_;

  // Workspace layout: [shw: B*HW floats][sel: B*M ints][zbuf: ZPAD_FLOATS floats]
  float* shw  = (float*)d_ws;
  int*   sel  = (int*)((char*)d_ws + (size_t)B_ * HW_ * sizeof(float));
  float* zbuf = (float*)((char*)d_ws + (size_t)B_ * HW_ * sizeof(float)
                                     + (size_t)B_ * M_ * sizeof(int));

  zero_pad_kernel<<<dim3(1), dim3(ZPAD_FLOATS), 0, stream>>>(zbuf);
  gemm_rowstats_kernel<<<dim3(B_ * (HW_ / 16)), dim3(64), 0, stream>>>(k, mk, zbuf, shw);
  rank_select_kernel<<<dim3(HW_ / 256, B_), dim3(256), 0, stream>>>(shw, sel);
  gather_kernel<<<dim3(P_, M_, B_), dim3(256), 0, stream>>>(k, v, sel, mk_out, mv_out);
}